// canny_edge_detection_76020921139353
// MI455X (gfx1250) — compile-verified
//
#include <hip/hip_runtime.h>
#include <stdint.h>

#define IMG_H 2048
#define IMG_W 2048
#define HW (IMG_H * IMG_W)
#define THRESH 10.0f

#define TILE 32
#define RAWD 40   // TILE + 8 (input halo: 2 blur + 2 blur + 1 sobel + 1 nms -> 4 each side)
#define HBW  36   // TILE + 4 (hblur cols)
#define BLD  36   // blurred region (TILE + 4)
#define GMD  34   // grad-mag region (TILE + 2)

typedef __attribute__((ext_vector_type(2))) float v2f;
typedef __attribute__((ext_vector_type(8))) float v8f;

// banded 5-tap weight: w[j] if j in [0,5) else 0 (branchless select chain)
__device__ __forceinline__ float band5(int j, float g0, float g1, float g2, float g3, float g4) {
    float w = 0.f;
    w = (j == 0) ? g0 : w;
    w = (j == 1) ? g1 : w;
    w = (j == 2) ? g2 : w;
    w = (j == 3) ? g3 : w;
    w = (j == 4) ? g4 : w;
    return w;
}

__global__ __launch_bounds__(256)
void canny_fused_kernel(const float* __restrict__ img, const float* __restrict__ gw,
                        float* __restrict__ out) {
    __shared__ __attribute__((aligned(16))) float s_raw[3][RAWD][RAWD]; // 19200 B, row = 160 B (16B-mult)
    __shared__ float s_hb[3][RAWD][HBW];                                // 17280 B
    __shared__ float s_bl[3][BLD][BLD];                                 // 15552 B
    __shared__ float s_gm[GMD * GMD];                                   //  4624 B

    const int tid  = threadIdx.x;
    const int lane = tid & 31;
    const int wid  = tid >> 5;
    const int x0 = blockIdx.x * TILE;
    const int y0 = blockIdx.y * TILE;

    const float g0 = gw[0], g1 = gw[1], g2 = gw[2], g3 = gw[3], g4 = gw[4];

    const bool interior = (blockIdx.x > 0) && (blockIdx.x < (IMG_W / TILE) - 1) &&
                          (blockIdx.y > 0) && (blockIdx.y < (IMG_H / TILE) - 1);

    // ---- Stage 1: raw halo tile -> LDS -------------------------------------
    if (interior) {
        // 3 ch x 40 rows x 10 float4 segments = 1200 async b128 DMA units.
        // Row starts: (x0-4)*4 bytes, x0 mult of 32 -> 16B aligned.
        for (int u = tid; u < 1200; u += 256) {
            int ch  = u / 400;
            int rem = u - ch * 400;
            int row = rem / 10;
            int seg = rem - row * 10;
            const float* src = img + ch * HW + (y0 - 4 + row) * IMG_W + (x0 - 4) + seg * 4;
            uint32_t dst = (uint32_t)(uintptr_t)&s_raw[ch][row][seg * 4];
            asm volatile("global_load_async_to_lds_b128 %0, %1, off"
                         :: "v"(dst), "v"(src) : "memory");
        }
        asm volatile("s_wait_asynccnt 0" ::: "memory");
    } else {
        // border: zero-filled scalar loads (reference zero-pads every conv)
        for (int u = tid; u < 3 * RAWD * RAWD; u += 256) {
            int ch  = u / (RAWD * RAWD);
            int rem = u - ch * (RAWD * RAWD);
            int yy  = rem / RAWD;
            int xx  = rem - yy * RAWD;
            int gy = y0 - 4 + yy, gx = x0 - 4 + xx;
            float v = 0.f;
            if (gy >= 0 && gy < IMG_H && gx >= 0 && gx < IMG_W)
                v = img[ch * HW + gy * IMG_W + gx];
            s_raw[ch][yy][xx] = v;
        }
    }
    __syncthreads();

    // ---- Stage 2: horizontal 5-tap blur (40 rows x 36 cols x 3 ch) ---------
    for (int u = tid; u < 3 * RAWD * HBW; u += 256) {
        int ch  = u / (RAWD * HBW);
        int rem = u - ch * (RAWD * HBW);
        int yy  = rem / HBW;
        int c   = rem - yy * HBW;
        const float* r = &s_raw[ch][yy][c];
        s_hb[ch][yy][c] = g0 * r[0] + g1 * r[1] + g2 * r[2] + g3 * r[3] + g4 * r[4];
    }
    __syncthreads();

    // ---- Stage 3: vertical blur via WMMA f32 16x16x4 -----------------------
    // D(16x16) = A(16x20 banded gw) x B(20x16 hblur rows), 5 chained K=4 WMMAs.
    // 9 overlapping 16x16 tiles (offsets {0,16,20}^2) cover the 36x36 region; x3 ch = 27.
    {
        const int m = lane & 15;            // A: row M / B,D: col N
        const int khalf = (lane >> 4) * 2;  // K sub-pair held by this half-wave
        for (int t = wid; t < 27; t += 8) { // uniform per-wave loop: EXEC all ones
            int ch = t / 9;
            int r9 = t - ch * 9;
            int ti = r9 / 3;
            int tj = r9 - ti * 3;
            int t0 = (ti == 0) ? 0 : ((ti == 1) ? 16 : 20);
            int c0 = (tj == 0) ? 0 : ((tj == 1) ? 16 : 20);
            v8f acc = {0.f, 0.f, 0.f, 0.f, 0.f, 0.f, 0.f, 0.f};
            #pragma unroll
            for (int cc = 0; cc < 5; ++cc) {
                int k = 4 * cc + khalf;
                v2f a, b;
                a.x = band5(k - m,     g0, g1, g2, g3, g4);
                a.y = band5(k + 1 - m, g0, g1, g2, g3, g4);
                b.x = s_hb[ch][t0 + k][c0 + m];
                b.y = s_hb[ch][t0 + k + 1][c0 + m];
                acc = __builtin_amdgcn_wmma_f32_16x16x4_f32(
                          false, a, false, b, (short)0, acc, false, false);
            }
            int rbase = t0 + ((lane >> 4) << 3);  // D layout: vgpr r -> M = r (+8 for hi half)
            #pragma unroll
            for (int r = 0; r < 8; ++r)
                s_bl[ch][rbase + r][c0 + m] = acc[r];
        }
    }
    __syncthreads();

    // border tiles: blurred outside the image must read as 0 for the next conv
    if (!interior) {
        for (int u = tid; u < 3 * BLD * BLD; u += 256) {
            int ch  = u / (BLD * BLD);
            int rem = u - ch * (BLD * BLD);
            int yy  = rem / BLD;
            int xx  = rem - yy * BLD;
            int gy = y0 - 2 + yy, gx = x0 - 2 + xx;
            if (gy < 0 || gy >= IMG_H || gx < 0 || gx >= IMG_W)
                s_bl[ch][yy][xx] = 0.f;
        }
        __syncthreads();
    }

    // ---- Stage 4: Sobel + channel-summed magnitude (34x34, zero-forced OOB) -
    for (int u = tid; u < GMD * GMD; u += 256) {
        int ym = u / GMD;
        int xm = u - ym * GMD;
        int gy = y0 - 1 + ym, gx = x0 - 1 + xm;
        float mag = 0.f;
        if (gy >= 0 && gy < IMG_H && gx >= 0 && gx < IMG_W) {
            #pragma unroll
            for (int ch = 0; ch < 3; ++ch) {
                const float (*bl)[BLD] = s_bl[ch];
                float a00 = bl[ym][xm],     a01 = bl[ym][xm + 1],     a02 = bl[ym][xm + 2];
                float a10 = bl[ym + 1][xm],                           a12 = bl[ym + 1][xm + 2];
                float a20 = bl[ym + 2][xm], a21 = bl[ym + 2][xm + 1], a22 = bl[ym + 2][xm + 2];
                float gxv = (a00 - a02) + 2.f * (a10 - a12) + (a20 - a22);
                float gyv = (a00 + 2.f * a01 + a02) - (a20 + 2.f * a21 + a22);
                mag += sqrtf(gxv * gxv + gyv * gyv);
            }
        }
        s_gm[u] = mag;
    }
    __syncthreads();

    // ---- Stage 5: orientation, NMS gather, thresholds, all 8 plane writes ---
    float* o_blur = out;                // [3][H][W]
    float* o_gm   = out + 3 * HW;
    float* o_ori  = out + 4 * HW;
    float* o_thin = out + 5 * HW;
    float* o_thr  = out + 6 * HW;
    float* o_ear  = out + 7 * HW;

    for (int i = tid; i < TILE * TILE; i += 256) {
        int y = i >> 5, x = i & 31;     // consecutive lanes -> consecutive x (coalesced)
        int gy = y0 + y, gx = x0 + x;
        int o = gy * IMG_W + gx;

        float gxs = 0.f, gys = 0.f;
        #pragma unroll
        for (int ch = 0; ch < 3; ++ch) {
            const float (*bl)[BLD] = s_bl[ch];
            o_blur[ch * HW + o] = bl[y + 2][x + 2];
            float a00 = bl[y + 1][x + 1], a01 = bl[y + 1][x + 2], a02 = bl[y + 1][x + 3];
            float a10 = bl[y + 2][x + 1],                         a12 = bl[y + 2][x + 3];
            float a20 = bl[y + 3][x + 1], a21 = bl[y + 3][x + 2], a22 = bl[y + 3][x + 3];
            gxs += (a00 - a02) + 2.f * (a10 - a12) + (a20 - a22);
            gys += (a00 + 2.f * a01 + a02) - (a20 + 2.f * a21 + a22);
        }
        float g = s_gm[(y + 1) * GMD + (x + 1)];

        float deg = atan2f(gys, gxs) * (180.0f / 3.14159f);
        float kf  = rintf((deg + 180.0f) * (1.0f / 45.0f));  // RNE, matches jnp.round
        float ori = kf * 45.0f;
        int ik = (int)kf;                 // 0..8
        int ip = ik & 7;
        int in_ = (ik + 4) & 7;
        // direction deltas packed as nibbles (value+1), idx0 at LSB:
        // dx: {1,1,0,-1,-1,-1,0,1}  dy: {0,1,1,1,0,-1,-1,-1}
        int dxp = (int)((0x21000122u >> (ip * 4)) & 0xF) - 1;
        int dyp = (int)((0x00012221u >> (ip * 4)) & 0xF) - 1;
        int dxn = (int)((0x21000122u >> (in_ * 4)) & 0xF) - 1;
        int dyn = (int)((0x00012221u >> (in_ * 4)) & 0xF) - 1;
        float pos = g - s_gm[(y + 1 + dyp) * GMD + (x + 1 + dxp)];
        float neg = g - s_gm[(y + 1 + dyn) * GMD + (x + 1 + dxn)];
        float thin = (fminf(pos, neg) > 0.f) ? g : 0.f;

        o_gm[o]   = g;
        o_ori[o]  = ori;
        o_thin[o] = thin;
        o_thr[o]  = (thin < THRESH) ? 0.f : thin;
        o_ear[o]  = (g < THRESH) ? 0.f : g;
    }
}

extern "C" void kernel_launch(void* const* d_in, const int* in_sizes, int n_in,
                              void* d_out, int out_size, void* d_ws, size_t ws_size,
                              hipStream_t stream) {
    (void)in_sizes; (void)n_in; (void)out_size; (void)d_ws; (void)ws_size;
    const float* img = (const float*)d_in[0];
    const float* gw  = (const float*)d_in[1];   // gw_h (5 floats); gw_v is identical
    float* out = (float*)d_out;
    dim3 grid(IMG_W / TILE, IMG_H / TILE);
    canny_fused_kernel<<<grid, dim3(256), 0, stream>>>(img, gw, out);
}